// EntmaxAlpha_53987738911351
// MI455X (gfx1250) — compile-verified
//
#include <hip/hip_runtime.h>
#include <math.h>

// entmax-bisect over rows of 1024 f32, one wave (32 lanes) per row,
// row held in 32 VGPRs/lane for all 51 passes. gfx1250 / wave32.
// Streaming I/O uses non-temporal (TH=NT) b128 ops: 512 MB touched once,
// > 192 MB L2, never re-read -> don't cache it.

#define ROW_D    1024
#define CHUNKS   8                 // 8 x (4 floats) x 32 lanes = 1024 floats
#define WAVES_PB 8
#define BLOCK_T  (WAVES_PB * 32)
#define N_ITER   50

typedef float v4f __attribute__((ext_vector_type(4)));

// XOR-butterfly lane exchange via ds_swizzle_b32 (group-of-32 mode):
// offset = {1'b0, and_mask=0x1f, or_mask=0, xor_mask=MASK} = 0x7C00 | MASK
template <int MASK>
__device__ __forceinline__ float xor_bfly(float v) {
  return __int_as_float(
      __builtin_amdgcn_ds_swizzle(__float_as_int(v), 0x7C00 | MASK));
}

__device__ __forceinline__ float wave_sum32(float v) {
  v += xor_bfly<16>(v);
  v += xor_bfly<8>(v);
  v += xor_bfly<4>(v);
  v += xor_bfly<2>(v);
  v += xor_bfly<1>(v);
  return v;   // all 32 lanes hold the total
}

__device__ __forceinline__ float wave_max32(float v) {
  v = fmaxf(v, xor_bfly<16>(v));
  v = fmaxf(v, xor_bfly<8>(v));
  v = fmaxf(v, xor_bfly<4>(v));
  v = fmaxf(v, xor_bfly<2>(v));
  v = fmaxf(v, xor_bfly<1>(v));
  return v;
}

// clip(t,0)^inv = exp2(inv * log2(max(t,0))); t<=0 -> log2 = -inf -> 0 (inv>0)
__device__ __forceinline__ float powi_pos(float t, float inv) {
  float tt = fmaxf(t, 0.0f);
  return __builtin_amdgcn_exp2f(inv * __builtin_amdgcn_logf(tt));
}

__global__ __launch_bounds__(BLOCK_T) void entmax_bisect_rows(
    const float* __restrict__ X, const float* __restrict__ alpha_raw,
    float* __restrict__ Y, int nrows, int H) {
  const int lane = threadIdx.x & 31;
  const int row  = blockIdx.x * WAVES_PB + (threadIdx.x >> 5);
  if (row >= nrows) return;   // whole wave exits uniformly

  // rows laid out as (b, h, q) with q-extent 1024 -> head = (row >> 10) % H
  int h = (row >> 10) % H;
  h = __builtin_amdgcn_readfirstlane(h);          // wave-uniform -> scalar load
  const float a_raw = alpha_raw[h];
  const float am1   = 0.01f + 0.98f * a_raw;      // (alpha - 1) in (0.01, 0.99)
  const float inv   = 1.0f / am1;

  const v4f* __restrict__ xin =
      reinterpret_cast<const v4f*>(X) + (size_t)row * (ROW_D / 4);
  v4f* __restrict__ yout =
      reinterpret_cast<v4f*>(Y) + (size_t)row * (ROW_D / 4);

  __builtin_prefetch(xin + lane, 0, 0);           // global_prefetch_b8 (gfx1250)

  // ---- load row once (NT); keep Xs = X*(alpha-1) resident in 32 VGPRs/lane --
  float xs[CHUNKS * 4];
#pragma unroll
  for (int c = 0; c < CHUNKS; ++c) {
    v4f v = __builtin_nontemporal_load(xin + c * 32 + lane); // b128, TH=NT
    xs[4 * c + 0] = v[0] * am1;
    xs[4 * c + 1] = v[1] * am1;
    xs[4 * c + 2] = v[2] * am1;
    xs[4 * c + 3] = v[3] * am1;
  }

  // ---- row max ----
  float m = xs[0];
#pragma unroll
  for (int i = 1; i < CHUNKS * 4; ++i) m = fmaxf(m, xs[i]);
  m = wave_max32(m);

  float tau_lo = m - 1.0f;
  // tau_hi - tau_lo = 1 - (1/1024)^inv = 1 - exp2(-10*inv)
  float dm = 1.0f - __builtin_amdgcn_exp2f(-10.0f * inv);

  // ---- f_lo (computed once, constant across iterations as in reference) ----
  float s0 = 0.f, s1 = 0.f, s2 = 0.f, s3 = 0.f;
#pragma unroll
  for (int c = 0; c < CHUNKS; ++c) {
    s0 += powi_pos(xs[4 * c + 0] - tau_lo, inv);
    s1 += powi_pos(xs[4 * c + 1] - tau_lo, inv);
    s2 += powi_pos(xs[4 * c + 2] - tau_lo, inv);
    s3 += powi_pos(xs[4 * c + 3] - tau_lo, inv);
  }
  const float f_lo = wave_sum32((s0 + s1) + (s2 + s3)) - 1.0f;

  // ---- 50 branchless bisection steps, row stays in registers ----
  float tau_m = tau_lo;
#pragma unroll 1
  for (int it = 0; it < N_ITER; ++it) {
    dm *= 0.5f;
    tau_m = tau_lo + dm;
    float t0 = 0.f, t1 = 0.f, t2 = 0.f, t3 = 0.f;
#pragma unroll
    for (int c = 0; c < CHUNKS; ++c) {
      t0 += powi_pos(xs[4 * c + 0] - tau_m, inv);
      t1 += powi_pos(xs[4 * c + 1] - tau_m, inv);
      t2 += powi_pos(xs[4 * c + 2] - tau_m, inv);
      t3 += powi_pos(xs[4 * c + 3] - tau_m, inv);
    }
    const float f_m = wave_sum32((t0 + t1) + (t2 + t3)) - 1.0f;
    tau_lo = (f_m * f_lo >= 0.0f) ? tau_m : tau_lo;   // v_cndmask, branchless
  }

  // ---- final p at last tau_m, then normalize to sum 1 ----
  s0 = s1 = s2 = s3 = 0.f;
#pragma unroll
  for (int c = 0; c < CHUNKS; ++c) {
    float p0 = powi_pos(xs[4 * c + 0] - tau_m, inv);
    float p1 = powi_pos(xs[4 * c + 1] - tau_m, inv);
    float p2 = powi_pos(xs[4 * c + 2] - tau_m, inv);
    float p3 = powi_pos(xs[4 * c + 3] - tau_m, inv);
    xs[4 * c + 0] = p0;  s0 += p0;
    xs[4 * c + 1] = p1;  s1 += p1;
    xs[4 * c + 2] = p2;  s2 += p2;
    xs[4 * c + 3] = p3;  s3 += p3;
  }
  const float ssum = wave_sum32((s0 + s1) + (s2 + s3));
  const float r = 1.0f / ssum;                     // one precise divide per row

#pragma unroll
  for (int c = 0; c < CHUNKS; ++c) {
    v4f o;
    o[0] = xs[4 * c + 0] * r;
    o[1] = xs[4 * c + 1] * r;
    o[2] = xs[4 * c + 2] * r;
    o[3] = xs[4 * c + 3] * r;
    __builtin_nontemporal_store(o, yout + c * 32 + lane);    // b128, TH=NT
  }
}

extern "C" void kernel_launch(void* const* d_in, const int* in_sizes, int n_in,
                              void* d_out, int out_size, void* d_ws, size_t ws_size,
                              hipStream_t stream) {
  const float* X     = (const float*)d_in[0];   // [B,H,Q,K] f32, K = 1024
  const float* alpha = (const float*)d_in[1];   // [H] f32
  float* Y           = (float*)d_out;

  const long long total = (long long)in_sizes[0];
  const int H           = in_sizes[1];
  const int nrows       = (int)(total / ROW_D);

  const int blocks = (nrows + WAVES_PB - 1) / WAVES_PB;
  entmax_bisect_rows<<<blocks, BLOCK_T, 0, stream>>>(X, alpha, Y, nrows, H);
}